// encoderLoss_14826227106538
// MI455X (gfx1250) — compile-verified
//
#include <hip/hip_runtime.h>

#define NSAMP 10000
#define BATCH 4096
#define DIM   512
#define MT    32     // output rows per block (2 WMMA M-tiles)
#define KT    16     // K elements staged per iteration
#define NT    64     // d-columns per wave (4 WMMA N-tiles)

typedef float v2f __attribute__((ext_vector_type(2)));
typedef float v8f __attribute__((ext_vector_type(8)));

// ---------------------------------------------------------------------------
// Kernel 0: per-row ||y_pred||^2  and per-block MSE partial sums.
// One wave (32 lanes) per row of 512 floats; fixed-order reductions only.
// ---------------------------------------------------------------------------
__global__ __launch_bounds__(256) void ysq_mse_kernel(
    const float* __restrict__ yp, const float* __restrict__ yt,
    float* __restrict__ ysq, float* __restrict__ msepart)
{
    __shared__ float wsum[8];
    const int lane = threadIdx.x & 31;
    const int wave = threadIdx.x >> 5;
    const int row  = blockIdx.x * 8 + wave;
    const float* pr = yp + (size_t)row * DIM;
    const float* tr = yt + (size_t)row * DIM;

    float sq = 0.f, ms = 0.f;
#pragma unroll
    for (int q = 0; q < 4; ++q) {
        const int c = q * 128 + lane * 4;
        float4 p = *(const float4*)(pr + c);
        float4 t = *(const float4*)(tr + c);
        sq += p.x*p.x + p.y*p.y + p.z*p.z + p.w*p.w;
        float dx = p.x - t.x, dy = p.y - t.y, dz = p.z - t.z, dw = p.w - t.w;
        ms += dx*dx + dy*dy + dz*dz + dw*dw;
    }
#pragma unroll
    for (int off = 16; off; off >>= 1) {
        sq += __shfl_xor(sq, off, 32);
        ms += __shfl_xor(ms, off, 32);
    }
    if (lane == 0) { ysq[row] = sq; wsum[wave] = ms; }
    __syncthreads();
    if (threadIdx.x == 0) {
        float s = 0.f;
        for (int w = 0; w < 8; ++w) s += wsum[w];
        msepart[blockIdx.x] = s;
    }
}

// ---------------------------------------------------------------------------
// Kernel 1: fused gather + (M @ Y) GEMM + quad_d/quad_m partials.
//   grid = (BATCH/MT, 2)  (y-dim selects mustlinks / cannotlinks)
//   block = 256 = 8 waves; wave w owns d-columns [w*64, w*64+64).
// Uses V_WMMA_F32_16X16X4_F32 (full fp32 matrix path).
// ---------------------------------------------------------------------------
__global__ __launch_bounds__(256) void quad_kernel(
    const float* __restrict__ yp,
    const int*   __restrict__ idx,
    const float* __restrict__ ml,
    const float* __restrict__ cl,
    const float* __restrict__ ysq,
    float* __restrict__ qdpart, float* __restrict__ qmpart)
{
    __shared__ float m_lds[MT * KT];      // 32x16 staged M patch
    __shared__ int   idxi[MT];
    __shared__ float degred[2 * 256];
    __shared__ float red[256];
    __shared__ float qdrow[MT];

    const int tid  = threadIdx.x;
    const int lane = tid & 31;
    const int wave = tid >> 5;
    const int bx   = blockIdx.x;          // row-tile index
    const int mat  = blockIdx.y;          // 0 = mustlinks, 1 = cannotlinks
    const int i0   = bx * MT;
    const float* __restrict__ L = mat ? cl : ml;

    if (tid < MT) idxi[tid] = idx[i0 + tid];
    __syncthreads();

    // gather coordinates for this thread: rows (gr, gr+16), column slot gc
    const int gr = tid >> 4;              // 0..15
    const int gc = tid & 15;              // 0..15
    const size_t rowA = (size_t)idxi[gr]      * NSAMP;
    const size_t rowB = (size_t)idxi[gr + 16] * NSAMP;

    float deg0 = 0.f, deg1 = 0.f;

    const v8f vzero = {0.f, 0.f, 0.f, 0.f, 0.f, 0.f, 0.f, 0.f};
    v8f acc[2][4];
#pragma unroll
    for (int m = 0; m < 2; ++m)
#pragma unroll
        for (int t = 0; t < 4; ++t) acc[m][t] = vzero;

    const int half    = lane >> 4;          // 0 or 1 (lane half)
    const int lan     = lane & 15;
    const int colbase = wave * NT + lan;    // + t*16 per N-tile

    for (int j = 0; j < BATCH; j += KT) {
        // ---- stage: gather 32x16 patch of M into LDS, fold row-degrees ----
        const int jc = idx[j + gc];
        const float m0 = L[rowA + jc];
        const float m1 = L[rowB + jc];
        m_lds[gr * KT + gc]        = m0;
        m_lds[(gr + 16) * KT + gc] = m1;
        deg0 += m0;
        deg1 += m1;
        __syncthreads();

        // ---- compute: 4 K-steps x 4 N-tiles x 2 M-tiles of 16x16x4 WMMA ---
#pragma unroll
        for (int kk = 0; kk < KT; kk += 4) {
            // A-frag layout: lanes 0-15 -> (K=kk,kk+1), lanes 16-31 -> (kk+2,kk+3)
            const float* a0p = &m_lds[lan * KT + kk + 2 * half];
            const float* a1p = &m_lds[(lan + 16) * KT + kk + 2 * half];
            const v2f a0 = { a0p[0], a0p[1] };
            const v2f a1 = { a1p[0], a1p[1] };
            const size_t rowoff = (size_t)(j + kk + 2 * half) * DIM;
#pragma unroll
            for (int t = 0; t < 4; ++t) {
                const int col = colbase + t * 16;
                const v2f b = { yp[rowoff + col], yp[rowoff + DIM + col] };
                acc[0][t] = __builtin_amdgcn_wmma_f32_16x16x4_f32(
                    false, a0, false, b, (short)0, acc[0][t], false, false);
                acc[1][t] = __builtin_amdgcn_wmma_f32_16x16x4_f32(
                    false, a1, false, b, (short)0, acc[1][t], false, false);
            }
        }
        __syncthreads();
    }

    // ---- quad_m partial: sum (M@Y) .* Y over this block's tile ------------
    float qm = 0.f;
#pragma unroll
    for (int m = 0; m < 2; ++m)
#pragma unroll
        for (int t = 0; t < 4; ++t)
#pragma unroll
            for (int r = 0; r < 8; ++r) {
                const int row = i0 + m * 16 + r + half * 8;   // D-tile M index
                const int col = colbase + t * 16;
                qm += acc[m][t][r] * yp[(size_t)row * DIM + col];
            }
    red[tid]          = qm;
    degred[tid]       = deg0;
    degred[256 + tid] = deg1;
    __syncthreads();

    // fixed-order tree reduction of qm over 256 threads (deterministic)
#pragma unroll
    for (int off = 128; off; off >>= 1) {
        if (tid < off) red[tid] += red[tid + off];
        __syncthreads();
    }

    // per-row degree reduction (fixed order), then deg_i * ||y_i||^2
    if (tid < MT) {
        const int base = (tid < 16) ? tid * 16 : 256 + (tid - 16) * 16;
        float d = 0.f;
#pragma unroll
        for (int c = 0; c < 16; ++c) d += degred[base + c];
        qdrow[tid] = d * ysq[i0 + tid];
    }
    __syncthreads();

    if (tid == 0) {
        float qd = 0.f;
        for (int r = 0; r < MT; ++r) qd += qdrow[r];
        qdpart[mat * gridDim.x + bx] = qd;
        qmpart[mat * gridDim.x + bx] = red[0];
    }
}

// ---------------------------------------------------------------------------
// Kernel 2: deterministic serial final reduction + loss assembly.
// ---------------------------------------------------------------------------
__global__ void final_kernel(const float* __restrict__ msepart,
                             const float* __restrict__ qd,
                             const float* __restrict__ qm,
                             float* __restrict__ out, int nmse, int nq)
{
    if (threadIdx.x == 0 && blockIdx.x == 0) {
        float mse = 0.f;
        for (int i = 0; i < nmse; ++i) mse += msepart[i];
        mse *= (1.0f / ((float)BATCH * (float)DIM));

        float qdm = 0.f, qmm = 0.f, qdc = 0.f, qmc = 0.f;
        for (int i = 0; i < nq; ++i) { qdm += qd[i];      qmm += qm[i];      }
        for (int i = 0; i < nq; ++i) { qdc += qd[nq + i]; qmc += qm[nq + i]; }

        const float scale = 2.0f / ((float)BATCH * (float)BATCH); // 2^-23, exact
        const float lm = (qdm - qmm) * scale;
        const float lc = (qdc - qmc) * scale;
        out[0] = mse + 0.5f * lm - 0.5f * lc;   // GAMMA = ALPHA = 0.5
        out[1] = lm;
        out[2] = lc;
    }
}

// ---------------------------------------------------------------------------
extern "C" void kernel_launch(void* const* d_in, const int* in_sizes, int n_in,
                              void* d_out, int out_size, void* d_ws, size_t ws_size,
                              hipStream_t stream)
{
    (void)in_sizes; (void)n_in; (void)out_size; (void)ws_size;
    const float* yp  = (const float*)d_in[0];   // y_pred      [B, D]
    const float* yt  = (const float*)d_in[1];   // y_truth     [B, D]
    const int*   idx = (const int*)  d_in[2];   // data_indx   [B]
    const float* ml  = (const float*)d_in[3];   // mustlinks   [N, N]
    const float* cl  = (const float*)d_in[4];   // cannotlinks [N, N]
    float* out = (float*)d_out;

    float* ws      = (float*)d_ws;
    float* ysq     = ws;                 // 4096 floats
    float* msepart = ws + 4096;          // 512 floats
    float* qdpart  = ws + 4608;          // 256 floats (2 mats x 128 blocks)
    float* qmpart  = ws + 4864;          // 256 floats

    ysq_mse_kernel<<<BATCH / 8, 256, 0, stream>>>(yp, yt, ysq, msepart);

    dim3 grid(BATCH / MT, 2);
    quad_kernel<<<grid, 256, 0, stream>>>(yp, idx, ml, cl, ysq, qdpart, qmpart);

    final_kernel<<<1, 32, 0, stream>>>(msepart, qdpart, qmpart, out,
                                       BATCH / 8, BATCH / MT);
}